// RotatedIoULoss_4724464025729
// MI455X (gfx1250) — compile-verified
//
#include <hip/hip_runtime.h>
#include <math.h>

#define EPSV 1e-6f

typedef __attribute__((ext_vector_type(2))) float v2f;
typedef __attribute__((ext_vector_type(8))) float v8f;

__device__ __forceinline__ float cross2(float ax, float ay, float bx, float by) {
    return ax * by - ay * bx;
}

// Monotone in atan2(y,x) over (-pi, pi] -> [-2, 2]; preserves argsort order.
__device__ __forceinline__ float pseudoangle(float x, float y) {
    float d = fabsf(x) + fabsf(y);
    if (d == 0.0f) return 0.0f;            // atan2(0,0) == 0
    return copysignf(1.0f - x / d, y);
}

__device__ __forceinline__ bool in_rect(float px, float py,
                                        const float* rx, const float* ry) {
    float ax = rx[0], ay = ry[0];
    float abx = rx[1] - ax, aby = ry[1] - ay;
    float adx = rx[3] - ax, ady = ry[3] - ay;
    float apx = px - ax, apy = py - ay;
    float dab = apx * abx + apy * aby;
    float dad = apx * adx + apy * ady;
    return (dab >= 0.0f) && (dab <= abx * abx + aby * aby) &&
           (dad >= 0.0f) && (dad <= adx * adx + ady * ady);
}

__global__ void __launch_bounds__(256)
riou_loss_stage1(const float* __restrict__ pred,
                 const float* __restrict__ target,
                 float* __restrict__ blockSums, int n) {
    __shared__ float sp[1280];
    __shared__ float st[1280];
    __shared__ float red[256];

    const int tid  = threadIdx.x;
    const int base = blockIdx.x * 1280;
    const int tot  = n * 5;

    // Coalesced staging of AoS (stride-5) data through LDS.
#pragma unroll
    for (int r = 0; r < 5; ++r) {
        int g = base + r * 256 + tid;
        float pv = (g < tot) ? pred[g]   : 0.0f;
        float tv = (g < tot) ? target[g] : 0.0f;
        sp[r * 256 + tid] = pv;
        st[r * 256 + tid] = tv;
    }
    __syncthreads();

    const int idx = blockIdx.x * 256 + tid;
    float loss = 0.0f;

    if (idx < n) {
        float b1[5], b2[5];
#pragma unroll
        for (int k = 0; k < 5; ++k) { b1[k] = sp[tid * 5 + k]; b2[k] = st[tid * 5 + k]; }

        // Corners (matches reference offset ordering).
        float c1x[4], c1y[4], c2x[4], c2y[4];
        {
            float ca = cosf(b1[4]), sa = sinf(b1[4]);
            float hw = b1[2] * 0.5f, hh = b1[3] * 0.5f;
            const float ox[4] = {-hw, hw, hw, -hw};
            const float oy[4] = {-hh, -hh, hh, hh};
#pragma unroll
            for (int i = 0; i < 4; ++i) {
                c1x[i] = b1[0] + ca * ox[i] - sa * oy[i];
                c1y[i] = b1[1] + sa * ox[i] + ca * oy[i];
            }
        }
        {
            float ca = cosf(b2[4]), sa = sinf(b2[4]);
            float hw = b2[2] * 0.5f, hh = b2[3] * 0.5f;
            const float ox[4] = {-hw, hw, hw, -hw};
            const float oy[4] = {-hh, -hh, hh, hh};
#pragma unroll
            for (int i = 0; i < 4; ++i) {
                c2x[i] = b2[0] + ca * ox[i] - sa * oy[i];
                c2y[i] = b2[1] + sa * ox[i] + ca * oy[i];
            }
        }

        float ptx[24], pty[24];
        bool  mk[24];

        // 16 edge-edge intersections.
#pragma unroll
        for (int i = 0; i < 4; ++i) {
            float px = c1x[i], py = c1y[i];
            float d1x = c1x[(i + 1) & 3] - px, d1y = c1y[(i + 1) & 3] - py;
#pragma unroll
            for (int j = 0; j < 4; ++j) {
                float qx = c2x[j], qy = c2y[j];
                float d2x = c2x[(j + 1) & 3] - qx, d2y = c2y[(j + 1) & 3] - qy;
                float qpx = qx - px, qpy = qy - py;
                float den = cross2(d1x, d1y, d2x, d2y);
                bool  ok  = fabsf(den) > 1e-10f;
                float dens = ok ? den : 1.0f;
                float t = cross2(qpx, qpy, d2x, d2y) / dens;
                float u = cross2(qpx, qpy, d1x, d1y) / dens;
                int k = i * 4 + j;
                ptx[k] = px + t * d1x;
                pty[k] = py + t * d1y;
                mk[k]  = ok && (t >= 0.0f) && (t <= 1.0f) && (u >= 0.0f) && (u <= 1.0f);
            }
        }
        // Corners of box1 inside box2, and vice versa.
#pragma unroll
        for (int j = 0; j < 4; ++j) {
            ptx[16 + j] = c1x[j]; pty[16 + j] = c1y[j];
            mk[16 + j]  = in_rect(c1x[j], c1y[j], c2x, c2y);
            ptx[20 + j] = c2x[j]; pty[20 + j] = c2y[j];
            mk[20 + j]  = in_rect(c2x[j], c2y[j], c1x, c1y);
        }

        // Centroid of masked points.
        float sx = 0.0f, sy = 0.0f, cnt = 0.0f;
#pragma unroll
        for (int k = 0; k < 24; ++k) {
            if (mk[k]) { sx += ptx[k]; sy += pty[k]; cnt += 1.0f; }
        }
        float inv = 1.0f / fmaxf(cnt, 1.0f);
        float cenx = sx * inv, ceny = sy * inv;

        // rel + sort key (pseudo-angle; masked -> 1e8 so it sorts last).
        float key[24];
#pragma unroll
        for (int k = 0; k < 24; ++k) {
            float rx = mk[k] ? (ptx[k] - cenx) : 0.0f;
            float ry = mk[k] ? (pty[k] - ceny) : 0.0f;
            ptx[k] = rx; pty[k] = ry;
            key[k] = mk[k] ? pseudoangle(rx, ry) : 1e8f;
        }

        // Stable adjacent compare-exchange network (bubble), fully in VGPRs.
#pragma unroll
        for (int pass = 0; pass < 23; ++pass) {
#pragma unroll
            for (int j = 0; j < 23; ++j) {
                if (j < 23 - pass) {
                    if (key[j] > key[j + 1]) {
                        float tk = key[j]; key[j] = key[j + 1]; key[j + 1] = tk;
                        float tx = ptx[j]; ptx[j] = ptx[j + 1]; ptx[j + 1] = tx;
                        float ty = pty[j]; pty[j] = pty[j + 1]; pty[j + 1] = ty;
                    }
                }
            }
        }

        // Shoelace over v = masked ? rel_sorted : p0 (p0 = first sorted rel).
        float p0x = ptx[0], p0y = pty[0];
        float vx[24], vy[24];
#pragma unroll
        for (int k = 0; k < 24; ++k) {
            bool valid = key[k] < 3.0f;   // valid keys are in [-2,2]; masked = 1e8
            vx[k] = valid ? ptx[k] : p0x;
            vy[k] = valid ? pty[k] : p0y;
        }
        float s2 = 0.0f;
#pragma unroll
        for (int k = 0; k < 24; ++k) {
            int kn = (k + 1) % 24;
            s2 += vx[k] * vy[kn] - vy[k] * vx[kn];
        }
        float inter = 0.5f * fabsf(s2);
        if (cnt < 3.0f) inter = 0.0f;

        float a1 = b1[2] * b1[3];
        float a2 = b2[2] * b2[3];
        float uni = fmaxf(a1 + a2 - inter, EPSV);
        float iou = fmaxf(inter / uni, EPSV);
        loss = -logf(iou);
    }

    // Deterministic in-block tree reduction.
    red[tid] = loss;
    __syncthreads();
#pragma unroll
    for (int s = 128; s > 0; s >>= 1) {
        if (tid < s) red[tid] += red[tid + s];
        __syncthreads();
    }
    if (tid == 0) blockSums[blockIdx.x] = red[0];
}

// Single wave32. Fixed-order accumulation, then v_wmma_f32_16x16x4_f32 with an
// all-ones A so D[m][n] = s_n + s_{n+16}; 4 shfl_xor steps finish the sum.
__global__ void __launch_bounds__(32)
riou_loss_stage2(const float* __restrict__ blockSums, int nb, float invN,
                 float* __restrict__ out) {
    int lane = threadIdx.x;
    float s = 0.0f;
    for (int i = lane; i < nb; i += 32) s += blockSums[i];

    v2f a = {1.0f, 1.0f};     // A = ones(16x4)
    v2f b = {s, 0.0f};        // B(4x16): column (lane%16) holds s, rest 0
    v8f c = {};
    v8f d = __builtin_amdgcn_wmma_f32_16x16x4_f32(
        /*neg_a=*/false, a, /*neg_b=*/false, b,
        /*c_mod=*/(short)0, c, /*reuse_a=*/false, /*reuse_b=*/false);

    float t = d[0];           // column sum replicated on every row
    t += __shfl_xor(t, 1, 32);
    t += __shfl_xor(t, 2, 32);
    t += __shfl_xor(t, 4, 32);
    t += __shfl_xor(t, 8, 32);

    if (lane == 0) out[0] = t * invN;   // LOSS_WEIGHT == 1.0
}

extern "C" void kernel_launch(void* const* d_in, const int* in_sizes, int n_in,
                              void* d_out, int out_size, void* d_ws, size_t ws_size,
                              hipStream_t stream) {
    const float* pred   = (const float*)d_in[0];
    const float* target = (const float*)d_in[1];
    int n  = in_sizes[0] / 5;
    int nb = (n + 255) / 256;
    float* blockSums = (float*)d_ws;

    riou_loss_stage1<<<nb, 256, 0, stream>>>(pred, target, blockSums, n);
    riou_loss_stage2<<<1, 32, 0, stream>>>(blockSums, nb, 1.0f / (float)n,
                                           (float*)d_out);
}